// _DepthwiseMultiScaleFIR_31877247271580
// MI455X (gfx1250) — compile-verified
//
#include <hip/hip_runtime.h>
#include <stdint.h>

// DepthwiseMultiScaleFIR: 4 causal depthwise 1D convs (k=3,7,15,31) over
// x[b,l,16,128] fp32. Memory-bound: ~0.67 GB moved => ~29us @ 23.3 TB/s.
// Strategy: stream x once per (b,t,c), keep 32-deep circular history in VGPRs,
// write 4 outputs with non-temporal stores (write-once data, keep L2 for the
// x halo). Stage per-block weight tile into LDS with CDNA5 async global->LDS
// DMA; prefetch next time tile.

#define NHEADS 16
#define HDIM   128
#define CCH    (NHEADS * HDIM)   // 2048 channels
#define LSEQ   4096
#define K0 3
#define K1 7
#define K2 15
#define K3 31
#define TCH 128                  // time steps per thread
#define BLK 256                  // threads per block (8 wave32), one channel each

// LDS float offsets for the 4 weight segments (channel-major slices)
#define OFF0 0
#define OFF1 (BLK * K0)                       // 768
#define OFF2 (OFF1 + BLK * K1)                // 2560
#define OFF3 (OFF2 + BLK * K2)                // 6400
#define SWTOT (OFF3 + BLK * K3)               // 14336 floats = 57,344 B LDS

#if defined(__HIP_DEVICE_COMPILE__) && defined(__gfx1250__)
#if __has_builtin(__builtin_amdgcn_global_load_async_to_lds_b32)
#define HAVE_ASYNC_LDS 1
#endif
#endif

typedef __attribute__((address_space(1))) int gas_int;   // global AS elem
typedef __attribute__((address_space(3))) int las_int;   // LDS AS elem

// Copy n contiguous floats global -> LDS, coalesced across the block.
__device__ __forceinline__ void stage_seg(const float* __restrict__ g,
                                          float* l, int n, int tid) {
#ifdef HAVE_ASYNC_LDS
  for (int i = tid; i < n; i += BLK) {
    __builtin_amdgcn_global_load_async_to_lds_b32(
        (gas_int*)(g + i), (las_int*)(l + i), /*offset=*/0, /*cpol=*/0);
  }
#else
  for (int i = tid; i < n; i += BLK) l[i] = g[i];
#endif
}

__global__ __launch_bounds__(BLK)
void _DepthwiseMultiScaleFIR_kernel(const float* __restrict__ x,
                                    const float* __restrict__ w0,
                                    const float* __restrict__ w1,
                                    const float* __restrict__ w2,
                                    const float* __restrict__ w3,
                                    float* __restrict__ out, int B) {
  __shared__ float sw[SWTOT];

  const int tid   = threadIdx.x;
  const int cBase = blockIdx.x * BLK;
  const int c     = cBase + tid;
  const int t0    = blockIdx.y * TCH;
  const int b     = blockIdx.z;

  // ---- Stage per-block weight tile into LDS (async DMA on CDNA5) ----
  stage_seg(w0 + (size_t)cBase * K0, &sw[OFF0], BLK * K0, tid);
  stage_seg(w1 + (size_t)cBase * K1, &sw[OFF1], BLK * K1, tid);
  stage_seg(w2 + (size_t)cBase * K2, &sw[OFF2], BLK * K2, tid);
  stage_seg(w3 + (size_t)cBase * K3, &sw[OFF3], BLK * K3, tid);
#ifdef HAVE_ASYNC_LDS
  asm volatile("s_wait_asynccnt 0" ::: "memory");  // own wave's DMAs landed
#endif
  __syncthreads();                                 // all waves' DMAs visible

  // ---- Pull this channel's taps into VGPRs, reversed so a[j] hits x[t-j].
  // XLA conv = cross-correlation with left pad (k-1): tap for x[t-j] is w[k-1-j].
  float a0[K0], a1[K1], a2[K2], a3[K3];
#pragma unroll
  for (int j = 0; j < K0; ++j) a0[j] = sw[OFF0 + tid * K0 + (K0 - 1 - j)];
#pragma unroll
  for (int j = 0; j < K1; ++j) a1[j] = sw[OFF1 + tid * K1 + (K1 - 1 - j)];
#pragma unroll
  for (int j = 0; j < K2; ++j) a2[j] = sw[OFF2 + tid * K2 + (K2 - 1 - j)];
#pragma unroll
  for (int j = 0; j < K3; ++j) a3[j] = sw[OFF3 + tid * K3 + (K3 - 1 - j)];

  // ---- Circular history: hist[t & 31] holds x at absolute time t.
  // t0 is a multiple of 32, so indices below are compile-time after unroll.
  float hist[32];
#pragma unroll
  for (int j = 1; j <= 31; ++j) {
    const int t = t0 - j;
    float v = 0.0f;
    if (t >= 0) v = x[((size_t)b * LSEQ + t) * CCH + c];
    hist[(32 - j) & 31] = v;
  }

  const size_t plane = (size_t)B * LSEQ * CCH;         // one output tensor
  const size_t base  = ((size_t)b * LSEQ + t0) * CCH + c;
  const float* xp = x + base;
  float* p0 = out + base;
  float* p1 = p0 + plane;
  float* p2 = p1 + plane;
  float* p3 = p2 + plane;

#pragma unroll 1
  for (int tt = 0; tt < TCH; tt += 32) {
    if (t0 + tt + 32 < LSEQ)                           // warm L2 for next tile
      __builtin_prefetch(xp + (size_t)(tt + 32) * CCH, 0, 1);
#pragma unroll
    for (int u = 0; u < 32; ++u) {
      const size_t off = (size_t)(tt + u) * CCH;
      const float xv = xp[off];                        // coalesced 128B/wave
      hist[u] = xv;                                    // (t0+tt+u) & 31 == u
      float y0 = 0.0f, y1 = 0.0f, y2 = 0.0f, y3 = 0.0f;
#pragma unroll
      for (int j = 0; j < K3; ++j) {
        const float h = (j == 0) ? xv : hist[(u - j) & 31];  // const index
        if (j < K0) y0 = fmaf(a0[j], h, y0);
        if (j < K1) y1 = fmaf(a1[j], h, y1);
        if (j < K2) y2 = fmaf(a2[j], h, y2);
        y3 = fmaf(a3[j], h, y3);
      }
      // Write-once streaming output: non-temporal stores keep the 192MB L2
      // free for x-halo reuse instead of 537MB of dead write data.
      __builtin_nontemporal_store(y0, p0 + off);
      __builtin_nontemporal_store(y1, p1 + off);
      __builtin_nontemporal_store(y2, p2 + off);
      __builtin_nontemporal_store(y3, p3 + off);
    }
  }
}

extern "C" void kernel_launch(void* const* d_in, const int* in_sizes, int n_in,
                              void* d_out, int out_size, void* d_ws, size_t ws_size,
                              hipStream_t stream) {
  const float* x  = (const float*)d_in[0];
  const float* w0 = (const float*)d_in[1];
  const float* w1 = (const float*)d_in[2];
  const float* w2 = (const float*)d_in[3];
  const float* w3 = (const float*)d_in[4];
  float* out = (float*)d_out;

  const int B = in_sizes[0] / (LSEQ * CCH);            // 4 for the reference
  dim3 grid(CCH / BLK, LSEQ / TCH, B);                 // (8, 32, 4)
  _DepthwiseMultiScaleFIR_kernel<<<grid, BLK, 0, stream>>>(x, w0, w1, w2, w3, out, B);
}